// LSTMGenerator_53154515255849
// MI455X (gfx1250) — compile-verified
//
#include <hip/hip_runtime.h>

typedef float v2f __attribute__((ext_vector_type(2)));
typedef float v8f __attribute__((ext_vector_type(8)));

// D = A(16x4 f32) * B(4x16 f32) + C(16x16 f32), full-precision CDNA5 WMMA
__device__ __forceinline__ v8f wmma4(v2f a, v2f b, v8f c) {
  return __builtin_amdgcn_wmma_f32_16x16x4_f32(false, a, false, b, (short)0, c,
                                               false, false);
}

// CDNA5 has native V_TANH_F32 (TRANS32 pipe); build sigmoid from it too.
__device__ __forceinline__ float tanhx(float v) {
#if __has_builtin(__builtin_amdgcn_tanhf)
  return __builtin_amdgcn_tanhf(v);
#else
  return 1.0f - 2.0f * __builtin_amdgcn_rcpf(1.0f + __expf(2.0f * v));
#endif
}
__device__ __forceinline__ float sigm(float v) {
#if __has_builtin(__builtin_amdgcn_tanhf)
  return fmaf(0.5f, __builtin_amdgcn_tanhf(0.5f * v), 0.5f);
#else
  return __builtin_amdgcn_rcpf(1.0f + __expf(-v));
#endif
}

// lane l <-> lane l^16 exchange, kept in the VALU pipe (v_permlanex16_b32)
__device__ __forceinline__ float xswap16(float v) {
#if __has_builtin(__builtin_amdgcn_permlanex16)
  int i = __float_as_int(v);
  i = __builtin_amdgcn_permlanex16(i, i, 0x76543210, 0xfedcba98, true, false);
  return __int_as_float(i);
#else
  return __shfl_xor(v, 16, 32);
#endif
}

// Orientation: gates^T = W_hh @ h^T  with C preloaded as x*W_ih + bias
//   A (constant) : W_hh tiles in A-layout   (lane = gate row)
//   B (dynamic)  : h^T chunks in B-layout   (lane = batch)
//   D            : gates^T                  (lane = batch, VGPR r -> hidden
//                                            index j = r + 8*half)
__global__ __launch_bounds__(32)
void lstm16_wmma(const float* __restrict__ x, const float* __restrict__ h0,
                 const float* __restrict__ c0, const float* __restrict__ W_ih,
                 const float* __restrict__ W_hh, const float* __restrict__ b_ih,
                 const float* __restrict__ b_hh, const float* __restrict__ W1,
                 const float* __restrict__ b1, const float* __restrict__ W2,
                 const float* __restrict__ b2, float* __restrict__ out, int T) {
  const int lane = threadIdx.x & 31;
  const int half = lane >> 4;   // K-pair select / D row-half select
  const int l    = lane & 15;   // A: row m ; B/D: column n (batch)
  const int koff = 2 * half;
  const int b0   = blockIdx.x * 16;

  // ---- recurrence A-matrix: 4 gate tiles (i,f,g,o) x 4 K-chunks of W_hh
  v2f Wr[4][4];
  // per-lane constants for accumulator init (row m = r + 8*half of each tile)
  float WiD[4][8], bD[4][8];
#pragma unroll
  for (int tl = 0; tl < 4; ++tl) {
    const int g = 16 * tl + l;  // A-layout gate row for this lane
#pragma unroll
    for (int kc = 0; kc < 4; ++kc)
      Wr[tl][kc] = *(const v2f*)&W_hh[g * 16 + 4 * kc + koff];
#pragma unroll
    for (int r = 0; r < 8; ++r) {
      const int gd = 16 * tl + r + 8 * half;  // D-layout gate row
      WiD[tl][r] = W_ih[gd];
      bD[tl][r]  = b_ih[gd] + b_hh[gd];
    }
  }

  // ---- head layer-1 A-matrix: z^T = W1 @ relu(h^T), b1 via accumulator init
  v2f H1[4];
#pragma unroll
  for (int kc = 0; kc < 4; ++kc)
    H1[kc] = *(const v2f*)&W1[l * 16 + 4 * kc + koff];
  float b1D[8], w2h[8];
#pragma unroll
  for (int r = 0; r < 8; ++r) {
    b1D[r] = b1[r + 8 * half];
    w2h[r] = W2[r + 8 * half];
  }
  const float b2v = b2[0];

  // ---- state: h^T as B-layout chunks (registers), c as D-layout (registers)
  v2f hB[4];
#pragma unroll
  for (int kc = 0; kc < 4; ++kc)
    hB[kc] = *(const v2f*)&h0[(b0 + l) * 16 + 4 * kc + koff];
  float c[8];  // c[b = l][j = r + 8*half] : contiguous -> b128 loads
  const float* crow = &c0[(long)(b0 + l) * 16 + 8 * half];
#pragma unroll
  for (int r = 0; r < 8; ++r) c[r] = crow[r];

  const float* xrow = x + (long)(b0 + l) * T;
  float* orow = out + (long)(b0 + l) * T;

  for (int t = 0; t < T; ++t) {
    const float xv = xrow[t];  // x[b] for this lane's batch column

    // accumulator init: x*W_ih + bias  (packed FMAs, off the WMMA chain)
    v8f G0, G1, G2, G3;
#pragma unroll
    for (int r = 0; r < 8; ++r) {
      G0[r] = fmaf(xv, WiD[0][r], bD[0][r]);
      G1[r] = fmaf(xv, WiD[1][r], bD[1][r]);
      G2[r] = fmaf(xv, WiD[2][r], bD[2][r]);
      G3[r] = fmaf(xv, WiD[3][r], bD[3][r]);
    }

    // gates^T += W_hh-tiles @ h^T   (4-deep WMMA chain per gate tile)
#pragma unroll
    for (int kc = 0; kc < 4; ++kc) {
      G0 = wmma4(Wr[0][kc], hB[kc], G0);
      G1 = wmma4(Wr[1][kc], hB[kc], G1);
      G2 = wmma4(Wr[2][kc], hB[kc], G2);
      G3 = wmma4(Wr[3][kc], hB[kc], G3);
    }

    // elementwise LSTM cell update (lane = batch, r -> hidden j = r + 8*half)
    float hD[8];
#pragma unroll
    for (int r = 0; r < 8; ++r) {
      const float ig = sigm(G0[r]);
      const float fg = sigm(G1[r]);
      const float gg = tanhx(G2[r]);
      const float og = sigm(G3[r]);
      const float cn = fg * c[r] + ig * gg;
      c[r]  = cn;
      hD[r] = og * tanhx(cn);
    }

    // rebuild h^T B-layout chunks: exchange 4 values with lane^16 (VALU)
    const float tA = xswap16(half ? hD[0] : hD[2]);
    const float tB = xswap16(half ? hD[1] : hD[3]);
    const float tC = xswap16(half ? hD[4] : hD[6]);
    const float tD = xswap16(half ? hD[5] : hD[7]);
    hB[0] = half ? (v2f){tA, tB}       : (v2f){hD[0], hD[1]};  // j 0,1 | 2,3
    hB[1] = half ? (v2f){tC, tD}       : (v2f){hD[4], hD[5]};  // j 4,5 | 6,7
    hB[2] = half ? (v2f){hD[2], hD[3]} : (v2f){tA, tB};        // j 8,9 |10,11
    hB[3] = half ? (v2f){hD[6], hD[7]} : (v2f){tC, tD};        // j12,13|14,15

    // head layer 1: z^T = W1 @ relu(h^T) + b1 (b1 as accumulator init)
    v8f Z;
#pragma unroll
    for (int r = 0; r < 8; ++r) Z[r] = b1D[r];
#pragma unroll
    for (int kc = 0; kc < 4; ++kc) {
      const v2f ya = {fmaxf(hB[kc][0], 0.f), fmaxf(hB[kc][1], 0.f)};
      Z = wmma4(H1[kc], ya, Z);
    }

    // head layer 2: out[b] = tanh( sum_n relu(z^T[n][b]) * W2[n] + b2 )
    float s = 0.f;
#pragma unroll
    for (int r = 0; r < 8; ++r) s = fmaf(fmaxf(Z[r], 0.f), w2h[r], s);
    s += xswap16(s);
    const float o = tanhx(s + b2v);
    if (!half) orow[t] = o;  // 16 lanes, one store instruction per step
  }
}

extern "C" void kernel_launch(void* const* d_in, const int* in_sizes, int n_in,
                              void* d_out, int out_size, void* d_ws,
                              size_t ws_size, hipStream_t stream) {
  const float* x    = (const float*)d_in[0];
  const float* h0   = (const float*)d_in[1];
  const float* c0   = (const float*)d_in[2];
  const float* W_ih = (const float*)d_in[3];
  const float* W_hh = (const float*)d_in[4];
  const float* b_ih = (const float*)d_in[5];
  const float* b_hh = (const float*)d_in[6];
  const float* W1   = (const float*)d_in[7];
  const float* b1   = (const float*)d_in[8];
  const float* W2   = (const float*)d_in[9];
  const float* b2   = (const float*)d_in[10];
  float* out = (float*)d_out;

  const int B = in_sizes[1] / 16;       // h0 is [B,16]
  const int T = in_sizes[0] / B;        // x  is [B,T,1]

  dim3 grid(B / 16), block(32);         // one wave per 16-batch tile
  lstm16_wmma<<<grid, block, 0, stream>>>(x, h0, c0, W_ih, W_hh, b_ih, b_hh,
                                          W1, b1, W2, b2, out, T);
}